// ES_21320217657885
// MI455X (gfx1250) — compile-verified
//
#include <hip/hip_runtime.h>
#include <cstdint>

#define DEVINL __device__ __forceinline__

constexpr int NS = 16384;        // series
constexpr int NT = 1024;         // time steps
constexpr int SS_ROW = NT + 12;  // 1036 cols in seasonalities output
constexpr int LD_ROW = NT - 1;   // 1023 cols in log-diff output
constexpr int TT = 32;           // time steps per tile
constexpr int TP = 36;           // padded LDS row (words) per series (mult of 4 -> b128 aligned)
constexpr int TILE_W = 32 * TP;  // 1152 words per tile buffer
constexpr int WAVE_W = 6 * TILE_W; // 3 y bufs + 3 out bufs per wave
constexpr int NWAVES = 2;
constexpr int BLOCK = NWAVES * 32;

// ---- CDNA5 async-copy primitives (ASYNCcnt path) ----------------------------

DEVINL void async_load_tile(const float* ysrc, unsigned lds_byte_base, int lane) {
  // tile = 32 series x 32 times; instr i: lane l -> series i*4 + l/8, times (l%8)*4..+3
  const int sl = lane >> 3;
  const int tl = (lane & 7) << 2;
  const unsigned long long gbase = (unsigned long long)(uintptr_t)ysrc;
#pragma unroll
  for (int i = 0; i < 8; ++i) {
    const unsigned voff = (unsigned)(((i * 4 + sl) * NT + tl) * 4);          // global byte off
    const unsigned lb   = lds_byte_base + (unsigned)(((i * 4 + sl) * TP + tl) * 4); // LDS byte
    asm volatile("global_load_async_to_lds_b128 %0, %1, %2"
                 :: "v"(lb), "v"(voff), "s"(gbase) : "memory");
  }
}

template <int N>
DEVINL void wait_async() {
  asm volatile("s_wait_asynccnt %0" :: "n"(N) : "memory");
}

// ---- recurrence over one tile (ring indices are compile-time constants) -----

template <int PHASE, int J0, int NSTEP>
DEVINL void compute_tile(float* sm, int yb, int olv, int oss, int old0,
                         float lev_sm, float cm, float (&ring)[12],
                         float& lev_prev, float& log_prev) {
#pragma unroll
  for (int i = 0; i < NSTEP; ++i) {
    const int j = J0 + i;
    const int r = (PHASE + i) % 12;
    const float yt = sm[yb + j];
    const float st = ring[r];
    const float nl  = fmaf(lev_sm, yt * __builtin_amdgcn_rcpf(st), cm * lev_prev);
    const float nsv = fmaf(lev_sm, yt * __builtin_amdgcn_rcpf(nl), cm * st);
    ring[r] = nsv;
    const float ln = __logf(nl);
    sm[olv + j] = nl;
    sm[oss + j] = nsv;
    sm[old0 + j] = ln - log_prev;
    log_prev = ln;
    lev_prev = nl;
  }
}

// ---- coalesced drain of the out-tiles ---------------------------------------

template <int J0>
DEVINL void drain_tile(const float* sm, int olv, int oss, int old0, int lane,
                       size_t s0w, int t0,
                       float* __restrict__ lv, float* __restrict__ ss,
                       float* __restrict__ ld) {
  if (J0 == 0 || lane >= J0) {
    const int t = t0 + lane;
#pragma unroll
    for (int i = 0; i < 32; ++i)
      lv[(s0w + i) * (size_t)NT + t] = sm[olv + i * TP + lane];
#pragma unroll
    for (int i = 0; i < 32; ++i)
      ss[(s0w + i) * (size_t)SS_ROW + t + 12] = sm[oss + i * TP + lane];
#pragma unroll
    for (int i = 0; i < 32; ++i)
      ld[(s0w + i) * (size_t)LD_ROW + t - 1] = sm[old0 + i * TP + lane];
  }
}

// ---- kernel -----------------------------------------------------------------

__global__ __launch_bounds__(BLOCK) void es_kernel(
    const float* __restrict__ y, const int* __restrict__ idxs,
    const float* __restrict__ lev_sms, const float* __restrict__ init_seas,
    float* __restrict__ lv, float* __restrict__ ss, float* __restrict__ ld) {
  __shared__ __align__(16) float sm[NWAVES * WAVE_W];

  const int tid  = threadIdx.x;
  const int lane = tid & 31;
  const int w    = tid >> 5;
  const int s    = blockIdx.x * BLOCK + tid;           // this lane's series
  const size_t s0w = (size_t)blockIdx.x * BLOCK + w * 32;

  const int ws = w * WAVE_W;
  const unsigned smem_base = (unsigned)(uintptr_t)&sm[0];
  const unsigned ybyte0 = smem_base + (unsigned)(ws * 4);
  const float* ywave = y + s0w * (size_t)NT;

  // prefetch tiles 0,1,2 (tile k loads times 32k..32k+31, 16B-aligned)
  async_load_tile(ywave + 0 * TT, ybyte0 + 0u * TILE_W * 4, lane);
  async_load_tile(ywave + 1 * TT, ybyte0 + 1u * TILE_W * 4, lane);
  async_load_tile(ywave + 2 * TT, ybyte0 + 2u * TILE_W * 4, lane);

  // ---- prologue (overlaps with async loads) ----
  const int id = idxs[s];
  const float xs = lev_sms[id];
  const float lev_sm = __builtin_amdgcn_rcpf(1.0f + __expf(-xs)); // sigmoid
  const float cm = 1.0f - lev_sm;

  float ring[12];
  float isv[12];
#pragma unroll
  for (int j = 0; j < 12; ++j) isv[j] = __expf(init_seas[(size_t)id * 12 + j]);
  const size_t ssrow = (size_t)s * SS_ROW;
#pragma unroll
  for (int j = 0; j < 12; ++j) ss[ssrow + j] = isv[j];
  ss[ssrow + 12] = isv[0];
#pragma unroll
  for (int j = 0; j < 12; ++j) ring[j] = isv[(j + 1) % 12];

  const float y0 = y[(size_t)s * NT];
  float lev_prev = y0 * __builtin_amdgcn_rcpf(isv[0]);
  lv[(size_t)s * NT] = lev_prev;
  float log_prev = __logf(lev_prev);

  const int rowoff = lane * TP;
  const int OLV = ws + 3 * TILE_W;
  const int OSS = ws + 4 * TILE_W;
  const int OLD = ws + 5 * TILE_W;

#define DO_TILE(PH, K, BUF, ISSUE_T, ISSUE_BUF, WAITN)                          \
  do {                                                                          \
    if ((ISSUE_T) >= 0)                                                         \
      async_load_tile(ywave + (ISSUE_T) * TT,                                   \
                      ybyte0 + (unsigned)((ISSUE_BUF) * TILE_W * 4), lane);     \
    wait_async<WAITN>();                                                        \
    compute_tile<PH, 0, 32>(sm, ws + (BUF) * TILE_W + rowoff, OLV + rowoff,     \
                            OSS + rowoff, OLD + rowoff, lev_sm, cm, ring,       \
                            lev_prev, log_prev);                                \
    drain_tile<0>(sm, OLV, OSS, OLD, lane, s0w, 32 * (K), lv, ss, ld);          \
  } while (0)

  // tile 0: steps t=1..31 (phase 0); tile 0's loads ready when asynccnt<=16
  wait_async<16>();
  compute_tile<0, 1, 31>(sm, ws + 0 * TILE_W + rowoff, OLV + rowoff,
                         OSS + rowoff, OLD + rowoff, lev_sm, cm, ring,
                         lev_prev, log_prev);
  drain_tile<1>(sm, OLV, OSS, OLD, lane, s0w, 0, lv, ss, ld);

  // tiles 1..27 (phase pattern 7,3,11), depth-2 prefetch
  for (int m = 0; m < 9; ++m) {
    const int k0 = 3 * m + 1;
    DO_TILE(7,  k0,     1, k0 + 2, 0, 16);
    DO_TILE(3,  k0 + 1, 2, k0 + 3, 1, 16);
    DO_TILE(11, k0 + 2, 0, k0 + 4, 2, 16);
  }
  // peeled tail
  DO_TILE(7,  28, 1, 30, 0, 16);
  DO_TILE(3,  29, 2, 31, 1, 16);
  DO_TILE(11, 30, 0, -1, 0, 8);
  DO_TILE(7,  31, 1, -1, 0, 0);
#undef DO_TILE
}

extern "C" void kernel_launch(void* const* d_in, const int* in_sizes, int n_in,
                              void* d_out, int out_size, void* d_ws, size_t ws_size,
                              hipStream_t stream) {
  const float* y         = (const float*)d_in[0];
  const int*   idxs      = (const int*)d_in[1];
  const float* lev_sms   = (const float*)d_in[2];
  const float* init_seas = (const float*)d_in[3];

  float* lv = (float*)d_out;
  float* ss = lv + (size_t)NS * NT;
  float* ld = ss + (size_t)NS * SS_ROW;

  dim3 grid(NS / BLOCK), block(BLOCK);
  hipLaunchKernelGGL(es_kernel, grid, block, 0, stream,
                     y, idxs, lev_sms, init_seas, lv, ss, ld);
}